// Block_71906342470306
// MI455X (gfx1250) — compile-verified
//
#include <hip/hip_runtime.h>
#include <math.h>
#include <stdint.h>

typedef float v2f __attribute__((ext_vector_type(2)));
typedef float v8f __attribute__((ext_vector_type(8)));
typedef unsigned int u32x4 __attribute__((ext_vector_type(4)));
typedef int i32x4 __attribute__((ext_vector_type(4)));
typedef int i32x8 __attribute__((ext_vector_type(8)));

#define D_MODEL 1024
#define N_STATE 16
#define E_INNER 2048
#define K_CONV  4
#define DT_RANK 64
#define BATCH   4
#define SEQ     1024
#define NTOK    (BATCH * SEQ)           /* 4096 tokens */
#define XZ_W    (2 * E_INNER)           /* 4096 */
#define XDB_W   (DT_RANK + 2 * N_STATE) /* 96 */

// ---------------------------------------------------------------------------
// TDM: issue a 2D tile load global -> LDS (D# per CDNA5 ISA)
//   tensor_d0/d1 : remaining tensor extent from tile start (OOB -> zero fill)
//   tile_d0/d1   : tile width (elems) / height (rows)
//   row_stride   : elements between rows (tensor_dim0_stride)
//   pad16        : insert 1 dword of LDS padding every 16 dwords (A-tile
//                  rows land at stride 17 -> conflict-free WMMA frag reads)
// Tracked by TENSORcnt; caller waits with s_wait_tensorcnt.
// ---------------------------------------------------------------------------
static __device__ __forceinline__ void tdm_tile_load(
    const void* gsrc, void* lds_dst,
    unsigned tensor_d0, unsigned tensor_d1,
    unsigned tile_d0, unsigned tile_d1,
    unsigned long long row_stride_elems, bool pad16)
{
    const unsigned long long ga = (unsigned long long)(uintptr_t)gsrc;
    const unsigned lds_off = (unsigned)(uintptr_t)lds_dst; // generic addr low 32 = LDS byte offset

    u32x4 g0;
    g0[0] = 1u;                                             // count=1 (valid, user mode)
    g0[1] = lds_off;                                        // lds_addr (bytes)
    g0[2] = (unsigned)(ga & 0xFFFFFFFFull);                 // global_addr[31:0]
    g0[3] = (unsigned)((ga >> 32) & 0x01FFFFFFull)          // global_addr[56:32]
          | (2u << 30);                                     // type = 2 ("image")

    unsigned w0 = 2u << 16;                                 // data_size = 4 bytes
    if (pad16) w0 |= (1u << 20) | (3u << 22);               // pad_enable, interval=16dw, amount=1dw
    i32x8 g1;
    g1[0] = (int)w0;
    g1[1] = (int)((tensor_d0 & 0xFFFFu) << 16);                               // tensor_dim0[15:0]
    g1[2] = (int)(((tensor_d0 >> 16) & 0xFFFFu) | ((tensor_d1 & 0xFFFFu) << 16));
    g1[3] = (int)(((tensor_d1 >> 16) & 0xFFFFu) | ((tile_d0 & 0xFFFFu) << 16));
    g1[4] = (int)(tile_d1 & 0xFFFFu);                                         // tile_dim1, tile_dim2=0
    g1[5] = (int)(unsigned)(row_stride_elems & 0xFFFFFFFFull);                // dim0_stride[31:0]
    g1[6] = (int)(unsigned)((row_stride_elems >> 32) & 0xFFFFull);            // dim0_stride[47:32]
    g1[7] = 0;                                                                // dim1_stride (unused, 2D)

    const i32x4 gz = {0, 0, 0, 0};
#if defined(__clang_major__) && __clang_major__ >= 23
    const i32x8 gz8 = {0, 0, 0, 0, 0, 0, 0, 0};
    __builtin_amdgcn_tensor_load_to_lds(g0, g1, gz, gz, gz8, 0);
#else
    __builtin_amdgcn_tensor_load_to_lds(g0, g1, gz, gz, 0);
#endif
}

// ---------------------------------------------------------------------------
// 1) RMSNorm: one block per token, 256 threads, float4 per thread (1024 elems)
// ---------------------------------------------------------------------------
__global__ __launch_bounds__(256) void rmsnorm_kernel(
    const float* __restrict__ x, const float* __restrict__ w,
    float* __restrict__ out)
{
    const int tok = blockIdx.x;
    const float4* xr = (const float4*)(x + (size_t)tok * D_MODEL);
    float4 v = xr[threadIdx.x];
    float ss = v.x * v.x + v.y * v.y + v.z * v.z + v.w * v.w;

    #pragma unroll
    for (int off = 16; off > 0; off >>= 1) ss += __shfl_xor(ss, off);

    __shared__ float red[8];
    if ((threadIdx.x & 31) == 0) red[threadIdx.x >> 5] = ss;
    __syncthreads();
    float tot = 0.f;
    #pragma unroll
    for (int i = 0; i < 8; ++i) tot += red[i];

    const float scale = rsqrtf(tot * (1.0f / D_MODEL) + 1e-6f);
    const float4 wv = ((const float4*)w)[threadIdx.x];
    float4 o;
    o.x = v.x * scale * wv.x;
    o.y = v.y * scale * wv.y;
    o.z = v.z * scale * wv.z;
    o.w = v.w * scale * wv.w;
    ((float4*)(out + (size_t)tok * D_MODEL))[threadIdx.x] = o;
}

// ---------------------------------------------------------------------------
// 2) f32 WMMA GEMM with double-buffered TDM tile staging.
//    C[M,N] = A[M,K] @ B[K,N] (+ optional add_src)
//    BM=128, BN=128, BK=16; 256 threads = 8 waves; wave tile 32x64.
//    Wave 0 drives the Tensor Data Mover (next tile) while all waves run
//    WMMAs on the current tile; sync = s_wait_tensorcnt + one barrier/K-step.
//    Requires: M % 128 == 0, K % 16 == 0. N: any multiple of 16 (TDM
//    zero-fills B cols >= N, stores guarded).
// ---------------------------------------------------------------------------
#define BM 128
#define BN 128
#define BK 16
#define A_LD 17   /* A-tile LDS row stride: 16 dwords + 1 TDM pad dword */

__global__ __launch_bounds__(256) void gemm_f32_wmma(
    const float* __restrict__ A, int lda,
    const float* __restrict__ B, int ldb,
    float* __restrict__ C, int ldc,
    int N, int K,
    const float* __restrict__ add_src)
{
    __shared__ float As[2][BM * A_LD];   // row-major [m][k], padded rows
    __shared__ float Bs[2][BK * BN];     // row-major [k][n]

    const int tid  = threadIdx.x;
    const int m0   = blockIdx.y * BM;
    const int n0   = blockIdx.x * BN;

    const int wave = tid >> 5;
    const int lane = tid & 31;
    const int half = lane >> 4;       // K-pair select / M-half select
    const int l16  = lane & 15;
    const int wm   = wave >> 1;       // 0..3 along M (32 rows each)
    const int wn   = wave & 1;        // 0..1 along N (64 cols each)
    const bool issuer = (wave == 0);  // one wave drives the TDM

    v8f acc[2][4];
    #pragma unroll
    for (int i = 0; i < 2; ++i)
        #pragma unroll
        for (int j = 0; j < 4; ++j) acc[i][j] = (v8f)0.0f;

    // prologue: stage tile 0 into buffer 0
    if (issuer) {
        tdm_tile_load(A + (size_t)m0 * lda, &As[0][0],
                      (unsigned)K, BM, BK, BM,
                      (unsigned long long)lda, true);
        tdm_tile_load(B + n0, &Bs[0][0],
                      (unsigned)(N - n0), (unsigned)K, BN, BK,
                      (unsigned long long)ldb, false);
    }

    int ib = 0;
    for (int kt = 0; kt < K; kt += BK, ib ^= 1) {
        if (issuer) __builtin_amdgcn_s_wait_tensorcnt(0); // current tile landed
        __syncthreads();                                  // publish to all waves

        const int ktn = kt + BK;
        if (issuer && ktn < K) {                          // prefetch next tile
            tdm_tile_load(A + (size_t)m0 * lda + ktn, &As[ib ^ 1][0],
                          (unsigned)(K - ktn), BM, BK, BM,
                          (unsigned long long)lda, true);
            tdm_tile_load(B + (size_t)ktn * ldb + n0, &Bs[ib ^ 1][0],
                          (unsigned)(N - n0), (unsigned)(K - ktn), BN, BK,
                          (unsigned long long)ldb, false);
        }

        const float* as = &As[ib][0];
        const float* bs = &Bs[ib][0];

        #pragma unroll
        for (int ks = 0; ks < BK; ks += 4) {
            v2f afr[2], bfr[4];
            #pragma unroll
            for (int i = 0; i < 2; ++i) {
                const int m = wm * 32 + i * 16 + l16;
                const int base = m * A_LD + ks + 2 * half;
                afr[i].x = as[base];
                afr[i].y = as[base + 1];
            }
            #pragma unroll
            for (int j = 0; j < 4; ++j) {
                const int n = wn * 64 + j * 16 + l16;
                bfr[j].x = bs[(ks + 2 * half + 0) * BN + n];
                bfr[j].y = bs[(ks + 2 * half + 1) * BN + n];
            }
            #pragma unroll
            for (int i = 0; i < 2; ++i)
                #pragma unroll
                for (int j = 0; j < 4; ++j)
                    acc[i][j] = __builtin_amdgcn_wmma_f32_16x16x4_f32(
                        false, afr[i], false, bfr[j], (short)0, acc[i][j],
                        false, false);
        }
        // no trailing barrier needed: next iteration's top barrier separates
        // this compute (buf ib) from the TDM write into buf ib (tile kt+2)
    }

    // store (C/D layout: VGPR jv -> M=jv (lanes 0-15) / M=8+jv (lanes 16-31))
    #pragma unroll
    for (int i = 0; i < 2; ++i) {
        #pragma unroll
        for (int j = 0; j < 4; ++j) {
            const int n = n0 + wn * 64 + j * 16 + l16;
            if (n < N) {
                #pragma unroll
                for (int jv = 0; jv < 8; ++jv) {
                    const int m = m0 + wm * 32 + i * 16 + half * 8 + jv;
                    const size_t idx = (size_t)m * ldc + n;
                    float cv = acc[i][j][jv];
                    if (add_src) cv += add_src[idx];
                    C[idx] = cv;
                }
            }
        }
    }
}

// ---------------------------------------------------------------------------
// 3) Causal depthwise conv (K=4) + SiLU: reads x_in = xz[:, 0:E], writes x_act
// ---------------------------------------------------------------------------
__global__ __launch_bounds__(256) void conv_silu_kernel(
    const float* __restrict__ xz, const float* __restrict__ conv_w,
    float* __restrict__ x_act)
{
    const size_t idx = (size_t)blockIdx.x * 256 + threadIdx.x; // B*L*E
    const int e = (int)(idx % E_INNER);
    const int l = (int)((idx / E_INNER) % SEQ);
    const int b = (int)(idx / ((size_t)E_INNER * SEQ));

    float s = 0.0f;
    #pragma unroll
    for (int k = 0; k < K_CONV; ++k) {
        const int li = l - (K_CONV - 1) + k;
        if (li >= 0)
            s += conv_w[e * K_CONV + k] *
                 xz[(size_t)(b * SEQ + li) * XZ_W + e];
    }
    x_act[idx] = s / (1.0f + expf(-s));   // silu
}

// ---------------------------------------------------------------------------
// 4) Selective scan: 1 lane per (b,e,n); 16-lane butterfly reduce for y.
//    Folds softplus(delta_pre + b_dt) and SiLU(z) gating.
//    Writes gated y into xz columns [0, E) (x_in is dead), h_last to d_out.
// ---------------------------------------------------------------------------
__global__ __launch_bounds__(256) void scan_kernel(
    const float* __restrict__ delta_pre,  // [NTOK][E]
    const float* __restrict__ x_act,      // [NTOK][E]
    const float* __restrict__ x_db,       // [NTOK][96]
    float* xz,                            // [NTOK][4096]: z in cols [E,2E), y out in [0,E)
    const float* __restrict__ A_log,      // [E][16]
    const float* __restrict__ b_dt,       // [E]
    const float* __restrict__ D_param,    // [E]
    float* __restrict__ h_last)           // [B][E][16]
{
    const int gid = blockIdx.x * 256 + threadIdx.x;   // B*E*16 = 131072
    const int n = gid & (N_STATE - 1);
    const int c = gid >> 4;                           // b*E + e
    const int e = c & (E_INNER - 1);
    const int b = c >> 11;

    const float An  = -expf(A_log[e * N_STATE + n]);
    const float bdt = b_dt[e];
    const float Dp  = D_param[e];
    float hn = 0.0f;

    const size_t tok0 = (size_t)b * SEQ;
    for (int l = 0; l < SEQ; ++l) {
        const size_t t = tok0 + l;
        float dpre = delta_pre[t * E_INNER + e] + bdt;
        const float delta = (dpre > 20.0f) ? dpre : log1pf(expf(dpre));
        const float u  = x_act[t * E_INNER + e];
        const float Bn = x_db[t * XDB_W + DT_RANK + n];
        const float Cn = x_db[t * XDB_W + DT_RANK + N_STATE + n];

        hn = expf(delta * An) * hn + (delta * u) * Bn;

        float part = hn * Cn;
        part += __shfl_xor(part, 8);
        part += __shfl_xor(part, 4);
        part += __shfl_xor(part, 2);
        part += __shfl_xor(part, 1);

        if (n == 0) {
            const float z = xz[t * XZ_W + E_INNER + e];
            const float y = part + u * Dp;
            xz[t * XZ_W + e] = y * (z / (1.0f + expf(-z)));
        }
    }
    h_last[(size_t)c * N_STATE + n] = hn;
}

// ---------------------------------------------------------------------------
// Launch
// ---------------------------------------------------------------------------
extern "C" void kernel_launch(void* const* d_in, const int* in_sizes, int n_in,
                              void* d_out, int out_size, void* d_ws, size_t ws_size,
                              hipStream_t stream)
{
    (void)in_sizes; (void)n_in; (void)out_size; (void)ws_size;

    const float* hidden  = (const float*)d_in[0]; // [4,1024,1024]
    const float* norm_w  = (const float*)d_in[1]; // [1024]
    const float* W_in    = (const float*)d_in[2]; // [1024,4096]
    const float* conv_w  = (const float*)d_in[3]; // [2048,1,4]
    const float* W_x     = (const float*)d_in[4]; // [2048,96]
    const float* W_dt    = (const float*)d_in[5]; // [64,2048]
    const float* b_dt    = (const float*)d_in[6]; // [2048]
    const float* A_log   = (const float*)d_in[7]; // [2048,16]
    const float* D_param = (const float*)d_in[8]; // [2048]
    const float* W_out   = (const float*)d_in[9]; // [2048,1024]

    float* out0   = (float*)d_out;                         // [4,1024,1024]
    float* h_last = out0 + (size_t)NTOK * D_MODEL;         // [4,2048,16]

    // workspace layout (floats)
    float* ws        = (float*)d_ws;
    float* x_norm    = ws;                                        // 4096*1024
    float* xz        = x_norm + (size_t)NTOK * D_MODEL;           // 4096*4096
    float* x_act     = xz + (size_t)NTOK * XZ_W;                  // 4096*2048
    float* x_db      = x_act + (size_t)NTOK * E_INNER;            // 4096*96
    float* delta_pre = x_db + (size_t)NTOK * XDB_W;               // 4096*2048

    // 1) RMSNorm
    rmsnorm_kernel<<<NTOK, 256, 0, stream>>>(hidden, norm_w, x_norm);

    // 2) xz = x_norm @ W_in   (M=4096, N=4096, K=1024)
    gemm_f32_wmma<<<dim3(XZ_W / BN, NTOK / BM), 256, 0, stream>>>(
        x_norm, D_MODEL, W_in, XZ_W, xz, XZ_W, XZ_W, D_MODEL, nullptr);

    // 3) depthwise causal conv + SiLU -> x_act
    conv_silu_kernel<<<((size_t)NTOK * E_INNER) / 256, 256, 0, stream>>>(
        xz, conv_w, x_act);

    // 4) x_db = x_act @ W_x   (M=4096, N=96, K=2048)  [TDM zero-fills N pad]
    gemm_f32_wmma<<<dim3(1, NTOK / BM), 256, 0, stream>>>(
        x_act, E_INNER, W_x, XDB_W, x_db, XDB_W, XDB_W, E_INNER, nullptr);

    // 5) delta_pre = x_db[:, :64] @ W_dt   (M=4096, N=2048, K=64)
    gemm_f32_wmma<<<dim3(E_INNER / BN, NTOK / BM), 256, 0, stream>>>(
        x_db, XDB_W, W_dt, E_INNER, delta_pre, E_INNER, E_INNER, DT_RANK, nullptr);

    // 6) selective scan (+ softplus, + SiLU(z) gating); y written into xz[:, :E]
    scan_kernel<<<(BATCH * E_INNER * N_STATE) / 256, 256, 0, stream>>>(
        delta_pre, x_act, x_db, xz, A_log, b_dt, D_param, h_last);

    // 7) out = y_gated @ W_out + residual  (M=4096, N=1024, K=2048)
    gemm_f32_wmma<<<dim3(D_MODEL / BN, NTOK / BM), 256, 0, stream>>>(
        xz, XZ_W, W_out, D_MODEL, out0, D_MODEL, D_MODEL, E_INNER, hidden);
}